// VanillaRNN_29798483100329
// MI455X (gfx1250) — compile-verified
//
#include <hip/hip_runtime.h>
#include <math.h>

#define Bb 64
#define Tt 4096
#define Dd 256
#define Hh 256
#define Oo 256

typedef __attribute__((ext_vector_type(2))) float v2f;
typedef __attribute__((ext_vector_type(8))) float v8f;

// D = A(16x4) * B(4x16) + C(16x16), fp32, wave32.
__device__ __forceinline__ v8f wmma_f32(v2f a, v2f b, v8f c) {
  return __builtin_amdgcn_wmma_f32_16x16x4_f32(
      /*neg_a=*/false, a, /*neg_b=*/false, b,
      /*c_mod=*/(short)0, c, /*reuse_a=*/false, /*reuse_b=*/false);
}

// ---------------------------------------------------------------------------
// Kernel 1: xp[t,b,n] = (x[b,t,:] @ W_xh)[n] + b_h[n]
// Block = 4 waves covering 4 n-tiles of ONE m-tile (A rows shared via WGP$).
// blockIdx.x (fastest) walks n-tile groups so all n-tiles of an A-row group
// are concurrent -> x streams from HBM once; W_xh (256 KB) stays in L2.
// ---------------------------------------------------------------------------
__global__ __launch_bounds__(128) void rnn_xproj(
    const float* __restrict__ x, const float* __restrict__ Wxh,
    const float* __restrict__ bh, float* __restrict__ xp) {
  const int lane = threadIdx.x & 31;
  const int wv   = threadIdx.x >> 5;
  const int half = lane >> 4;        // 0: K+0..1, 1: K+2..3
  const int l15  = lane & 15;
  const int koff = half * 2;

  const int  n0 = (blockIdx.x * 4 + wv) * 16;   // 16 n-tiles via 4 blocks x 4 waves
  const long r0 = (long)blockIdx.y * 16;        // m-tile: rows r0..r0+15 of (B*T, D)

  // B fragments: W_xh[:, n0:n0+16], 64 chunks of K=4
  v2f bw[64];
#pragma unroll
  for (int k = 0; k < 64; ++k) {
    const float* wp = Wxh + (size_t)(4 * k + koff) * Hh + n0 + l15;
    bw[k].x = wp[0];
    bw[k].y = wp[Hh];
  }

  const float* ap = x + (size_t)(r0 + l15) * Dd + koff;
  v8f c0 = {}, c1 = {};
#pragma unroll
  for (int k = 0; k < 64; k += 2) {
    v2f a0 = { ap[4 * k],     ap[4 * k + 1] };
    v2f a1 = { ap[4 * k + 4], ap[4 * k + 5] };
    c0 = wmma_f32(a0, bw[k],     c0);
    c1 = wmma_f32(a1, bw[k + 1], c1);
  }
  v8f c = c0 + c1;

  const float bias = bh[n0 + l15];
#pragma unroll
  for (int i = 0; i < 8; ++i) {
    const long r = r0 + i + 8 * half;    // global row = b*T + t
    const int  b = (int)(r >> 12);       // / T
    const int  t = (int)(r & (Tt - 1));
    xp[(size_t)t * (Bb * Hh) + (size_t)b * Hh + n0 + l15] = c[i] + bias;
  }
}

// ---------------------------------------------------------------------------
// Kernel 2: sequential scan, batch-parallel (no inter-WG sync needed).
// 4 workgroups x 16 waves; wave wv owns H-columns [16*wv, 16*wv+16).
// W_hh fragments persist in registers; h double-buffered in LDS (32 KB);
// next step's xp tile is register-prefetched so HBM latency hides behind
// the 64-WMMA chain.  hs overwrites xp in place.
// ---------------------------------------------------------------------------
__global__ __launch_bounds__(512) void rnn_recur(
    const float* __restrict__ h0, const float* __restrict__ Whh,
    float* __restrict__ xp /* in: xp, out: hs */,
    float* __restrict__ hfinal) {
  __shared__ float hbuf[2][16 * Hh];

  const int lane = threadIdx.x & 31;
  const int wv   = threadIdx.x >> 5;   // 0..15 -> column tile
  const int half = lane >> 4;
  const int l15  = lane & 15;
  const int koff = half * 2;
  const int n0   = wv * 16;
  const int b0   = blockIdx.x * 16;    // batch rows b0..b0+15

  // Persistent B fragments: W_hh[:, n0:n0+16]
  v2f bw[64];
#pragma unroll
  for (int k = 0; k < 64; ++k) {
    const float* wp = Whh + (size_t)(4 * k + koff) * Hh + n0 + l15;
    bw[k].x = wp[0];
    bw[k].y = wp[Hh];
  }

  // h_{-1} = h0
  for (int i = threadIdx.x; i < 16 * Hh; i += 512)
    hbuf[0][i] = h0[(size_t)(b0 + (i >> 8)) * Hh + (i & 255)];

  // Prefetch xp tile for t = 0
  v8f nextc;
#pragma unroll
  for (int i = 0; i < 8; ++i)
    nextc[i] = xp[(size_t)(b0 + i + 8 * half) * Hh + n0 + l15];

  __syncthreads();

  for (int t = 0; t < Tt; ++t) {
    const float* src = hbuf[t & 1];
    float*       dst = hbuf[(t + 1) & 1];
    float*       xpt = xp + (size_t)t * (Bb * Hh);

    // C starts from the prefetched input projection
    v8f c0 = nextc, c1 = {};

    // Issue prefetch of xp[t+1] now; its wait lands after the WMMA chain.
    const float* xnx = xp + (size_t)(t + 1 < Tt ? t + 1 : t) * (Bb * Hh);
    v8f pf;
#pragma unroll
    for (int i = 0; i < 8; ++i)
      pf[i] = xnx[(size_t)(b0 + i + 8 * half) * Hh + n0 + l15];

    const float* arow = src + l15 * Hh + koff;
#pragma unroll
    for (int k = 0; k < 64; k += 2) {
      v2f a0 = { arow[4 * k],     arow[4 * k + 1] };
      v2f a1 = { arow[4 * k + 4], arow[4 * k + 5] };
      c0 = wmma_f32(a0, bw[k],     c0);
      c1 = wmma_f32(a1, bw[k + 1], c1);
    }
    v8f c = c0 + c1;
    nextc = pf;

#pragma unroll
    for (int i = 0; i < 8; ++i) {
      const float h = tanhf(c[i]);
      const int   m = i + 8 * half;
      dst[m * Hh + n0 + l15]                = h;  // next-step state
      xpt[(size_t)(b0 + m) * Hh + n0 + l15] = h;  // hs[t] (in place)
    }
    __syncthreads();   // one barrier per timestep
  }

  // h_final = h_{T-1}  (lives in hbuf[T & 1])
  const float* hf = hbuf[Tt & 1];
  for (int i = threadIdx.x; i < 16 * Hh; i += 512)
    hfinal[(size_t)(b0 + (i >> 8)) * Hh + (i & 255)] = hf[i];
}

// ---------------------------------------------------------------------------
// Kernel 3: out[b,t,o] = (hs[t,b,:] @ W_hy)[o] + b_y[o]
// Same traversal as kernel 1: n-tiles innermost for single-pass hs streaming.
// ---------------------------------------------------------------------------
__global__ __launch_bounds__(128) void rnn_outproj(
    const float* __restrict__ hs, const float* __restrict__ Why,
    const float* __restrict__ by, float* __restrict__ out) {
  const int lane = threadIdx.x & 31;
  const int wv   = threadIdx.x >> 5;
  const int half = lane >> 4;
  const int l15  = lane & 15;
  const int koff = half * 2;

  const int  n0 = (blockIdx.x * 4 + wv) * 16;
  const long r0 = (long)blockIdx.y * 16;       // out row = b*T + t
  const int  b  = (int)(r0 >> 12);
  const int  t0 = (int)(r0 & (Tt - 1));        // 16 consecutive t, same b

  v2f bw[64];
#pragma unroll
  for (int k = 0; k < 64; ++k) {
    const float* wp = Why + (size_t)(4 * k + koff) * Oo + n0 + l15;
    bw[k].x = wp[0];
    bw[k].y = wp[Oo];
  }

  const float* ap = hs + (size_t)(t0 + l15) * (Bb * Hh) + (size_t)b * Hh + koff;
  v8f c0 = {}, c1 = {};
#pragma unroll
  for (int k = 0; k < 64; k += 2) {
    v2f a0 = { ap[4 * k],     ap[4 * k + 1] };
    v2f a1 = { ap[4 * k + 4], ap[4 * k + 5] };
    c0 = wmma_f32(a0, bw[k],     c0);
    c1 = wmma_f32(a1, bw[k + 1], c1);
  }
  v8f c = c0 + c1;

  const float bias = by[n0 + l15];
#pragma unroll
  for (int i = 0; i < 8; ++i) {
    const long r = r0 + i + 8 * half;
    out[(size_t)r * Oo + n0 + l15] = c[i] + bias;
  }
}

// ---------------------------------------------------------------------------
extern "C" void kernel_launch(void* const* d_in, const int* in_sizes, int n_in,
                              void* d_out, int out_size, void* d_ws, size_t ws_size,
                              hipStream_t stream) {
  const float* x   = (const float*)d_in[0];
  const float* h0  = (const float*)d_in[1];
  const float* Wxh = (const float*)d_in[2];
  const float* Whh = (const float*)d_in[3];
  const float* bh  = (const float*)d_in[4];
  const float* Why = (const float*)d_in[5];
  const float* by  = (const float*)d_in[6];

  float* out    = (float*)d_out;                      // (B,T,O)
  float* hfinal = out + (size_t)Bb * Tt * Oo;         // (B,H)
  float* xp     = (float*)d_ws;                       // (T,B,H), reused as hs

  // n-tile groups innermost (x), m-tiles outer (y) for L2-friendly streaming
  dim3 g1((Hh / 16) / 4, Bb * Tt / 16);
  rnn_xproj<<<g1, 128, 0, stream>>>(x, Wxh, bh, xp);

  rnn_recur<<<Bb / 16, 512, 0, stream>>>(h0, Whh, xp, hfinal);

  dim3 g3((Oo / 16) / 4, Bb * Tt / 16);
  rnn_outproj<<<g3, 128, 0, stream>>>(xp, Why, by, out);
}